// SpatioConvLayer_15504831938948
// MI455X (gfx1250) — compile-verified
//
#include <hip/hip_runtime.h>
#include <math.h>

typedef float v2f __attribute__((ext_vector_type(2)));
typedef float v8f __attribute__((ext_vector_type(8)));

#define B_ 8
#define C_ 64
#define N_ 400
#define L_ 12
#define H_ 4
#define NEG_INF_F (-9000000000000000.0f)
#define ALPHA_F 0.2f

#define MA_STRIDE 401   // 16 x 401 mean-att tile (bank-conflict-free A reads)
#define XS_STRIDE 80    // 400 x 80 x-slice (bank-conflict-free B reads)

// ---------------------------------------------------------------------------
// Kernel 1: Wa1[h,c] = sum_d W[h,c,d]*a[h,d],  Wa2[h,c] = sum_d W[h,c,d]*a[h,C+d]
// Folds the h = x*W einsum into two 256-float vectors (h never materialized).
// ---------------------------------------------------------------------------
__global__ void wa_kernel(const float* __restrict__ W, const float* __restrict__ a,
                          float* __restrict__ wa) {
    int t = threadIdx.x;                 // 256 threads: (h, c)
    int h = t >> 6, c = t & 63;
    float acc1 = 0.f, acc2 = 0.f;
    for (int d = 0; d < C_; ++d) {
        float wv = W[(h * C_ + c) * C_ + d];
        acc1 = fmaf(wv, a[h * 2 * C_ + d], acc1);
        acc2 = fmaf(wv, a[h * 2 * C_ + C_ + d], acc2);
    }
    wa[t]       = acc1;
    wa[256 + t] = acc2;
}

// ---------------------------------------------------------------------------
// Kernel 2: s1[h,b,l,v] = sum_c x[b,c,v,l]*Wa1[h,c]  (and s2 with Wa2)
// One thread per (b,v,l); x element loaded once, feeds all 8 accumulators.
// ---------------------------------------------------------------------------
__global__ __launch_bounds__(256) void s_kernel(const float* __restrict__ x,
                                                const float* __restrict__ wa,
                                                float* __restrict__ s1,
                                                float* __restrict__ s2) {
    __shared__ float sW1[H_ * C_];
    __shared__ float sW2[H_ * C_];
    int tid = threadIdx.x;
    sW1[tid] = wa[tid];
    sW2[tid] = wa[256 + tid];
    __syncthreads();

    int gid = blockIdx.x * 256 + tid;    // < B*N*L = 38400 exactly
    int b   = gid / (N_ * L_);
    int rem = gid - b * (N_ * L_);
    int v   = rem / L_;
    int l   = rem - v * L_;

    float a1[H_] = {0.f, 0.f, 0.f, 0.f};
    float a2[H_] = {0.f, 0.f, 0.f, 0.f};
    const float* xp = x + ((size_t)b * C_ * N_ + v) * L_ + l;
    for (int c = 0; c < C_; ++c) {
        float xv = xp[(size_t)c * N_ * L_];
#pragma unroll
        for (int h = 0; h < H_; ++h) {
            a1[h] = fmaf(xv, sW1[h * C_ + c], a1[h]);
            a2[h] = fmaf(xv, sW2[h * C_ + c], a2[h]);
        }
    }
#pragma unroll
    for (int h = 0; h < H_; ++h) {
        int o = ((h * B_ + b) * L_ + l) * N_ + v;
        s1[o] = a1[h];
        s2[o] = a2[h];
    }
}

// ---------------------------------------------------------------------------
// Kernel 3: logAdj[i,j] = log(adj + 1e-8), reused by all 96*(25) blocks.
// ---------------------------------------------------------------------------
__global__ void logadj_kernel(const float* __restrict__ adj, float* __restrict__ lg) {
    int t = blockIdx.x * 256 + threadIdx.x;
    if (t < N_ * N_) lg[t] = logf(adj[t] + 1e-8f);
}

// ---------------------------------------------------------------------------
// Kernel 4 (fused): per (b, l, 16-row i-tile):
//   scores -> softmax(j) per head -> mean over heads (all in LDS)
//   then out[i-tile, c=0..63] = mean_att(16x400) @ xT(400x64) via
//   v_wmma_f32_16x16x4_f32 (100 K-steps, 1 c-tile of 16 per wave), ELU, store.
// LDS budget: 16*401 (meanAtt) + max(score-phase 8.2K floats, 400*80 x-slice)
//           = 38416 floats = 150.1 KB of the 320 KB/WGP.
// ---------------------------------------------------------------------------
__global__ __launch_bounds__(128) void fused_att_kernel(
    const float* __restrict__ x, const float* __restrict__ adj,
    const float* __restrict__ s1g, const float* __restrict__ s2g,
    const float* __restrict__ logAdj, float* __restrict__ out) {
    extern __shared__ float smem[];
    const int tid = threadIdx.x;
    const int i0  = blockIdx.x * 16;   // i-tile
    const int l   = blockIdx.y;
    const int b   = blockIdx.z;

    float* sMA     = smem;                 // 16 x MA_STRIDE
    float* sU      = smem + 16 * MA_STRIDE;
    float* sS1     = sU;                   // H*400
    float* sS2     = sU + 1600;            // H*16
    float* sRed    = sU + 1664;            // 128
    float* sRowMax = sU + 1792;            // 16
    float* sRowSum = sU + 1808;            // 16
    float* sScore  = sU + 1824;            // 16*400
    float* sXs     = sU;                   // 400 x XS_STRIDE (phase C, aliases above)

    for (int t = tid; t < 16 * MA_STRIDE; t += 128) sMA[t] = 0.f;
    for (int t = tid; t < H_ * N_; t += 128) {
        int h = t / N_, j = t - h * N_;
        sS1[t] = s1g[((h * B_ + b) * L_ + l) * N_ + j];
    }
    for (int t = tid; t < H_ * 16; t += 128) {
        int h = t >> 4, ii = t & 15;
        sS2[t] = s2g[((h * B_ + b) * L_ + l) * N_ + i0 + ii];
    }
    __syncthreads();

    const int row = tid >> 3;   // 0..15 : i within tile
    const int sub = tid & 7;    // 8 threads cooperate per row

    for (int h = 0; h < H_; ++h) {
        // scores for this head
        for (int t = tid; t < 16 * N_; t += 128) {
            int ii = t / N_, j = t - ii * N_;
            float av = adj[(i0 + ii) * N_ + j];
            float sc;
            if (av == 0.f) {
                sc = NEG_INF_F;
            } else {
                float z = sS2[h * 16 + ii] + sS1[h * N_ + j];
                float e = (z >= 0.f) ? z : (ALPHA_F * z);   // leaky_relu
                sc = e * logAdj[(i0 + ii) * N_ + j];
            }
            sScore[t] = sc;
        }
        __syncthreads();
        // row max
        float m = NEG_INF_F;
        for (int k = 0; k < N_ / 8; ++k)
            m = fmaxf(m, sScore[row * N_ + sub + k * 8]);
        sRed[row * 8 + sub] = m;
        __syncthreads();
        if (tid < 16) {
            float mm = sRed[tid * 8];
            for (int s = 1; s < 8; ++s) mm = fmaxf(mm, sRed[tid * 8 + s]);
            sRowMax[tid] = mm;
        }
        __syncthreads();
        // exp (stored in place) + row sum
        float rm = sRowMax[row];
        float sacc = 0.f;
        for (int k = 0; k < N_ / 8; ++k) {
            int idx = row * N_ + sub + k * 8;
            float ex = __expf(sScore[idx] - rm);
            sScore[idx] = ex;
            sacc += ex;
        }
        sRed[row * 8 + sub] = sacc;
        __syncthreads();
        if (tid < 16) {
            float ss = 0.f;
            for (int s = 0; s < 8; ++s) ss += sRed[tid * 8 + s];
            sRowSum[tid] = ss;
        }
        __syncthreads();
        // accumulate head-mean attention
        for (int t = tid; t < 16 * N_; t += 128) {
            int ii = t / N_, j = t - ii * N_;
            sMA[ii * MA_STRIDE + j] += sScore[t] * (0.25f / sRowSum[ii]);
        }
        __syncthreads();
    }

    // stage x slice (b, :, :, l) into LDS as xs[j][c], padded stride
    for (int t = tid; t < C_ * N_; t += 128) {
        int c = t / N_, j = t - c * N_;
        sXs[j * XS_STRIDE + c] = x[(((size_t)b * C_ + c) * N_ + j) * L_ + l];
    }
    __syncthreads();

    // WMMA f32 16x16x4: D[i, c] = sum_j MA[i][j] * xs[j][c]
    const int wave = tid >> 5;
    const int lane = tid & 31;
    const int half = lane >> 4;       // K group: lanes 0-15 -> K=0,1 ; 16-31 -> K=2,3
    const int mrow = lane & 15;
    const int cw   = wave * 16;       // c-tile per wave

    v8f acc = {};
    for (int jk = 0; jk < N_ / 4; ++jk) {
        int j0 = jk * 4 + half * 2;
        v2f Af, Bf;
        Af.x = sMA[mrow * MA_STRIDE + j0];
        Af.y = sMA[mrow * MA_STRIDE + j0 + 1];
        Bf.x = sXs[(j0    ) * XS_STRIDE + cw + mrow];
        Bf.y = sXs[(j0 + 1) * XS_STRIDE + cw + mrow];
        acc = __builtin_amdgcn_wmma_f32_16x16x4_f32(
            /*neg_a=*/false, Af, /*neg_b=*/false, Bf,
            /*c_mod=*/(short)0, acc, /*reuse_a=*/false, /*reuse_b=*/false);
    }

    // D layout: VGPR r -> M = r (lanes 0-15) / r+8 (lanes 16-31), N = lane&15
#pragma unroll
    for (int r = 0; r < 8; ++r) {
        int i = i0 + r + half * 8;
        int c = cw + mrow;
        float v = acc[r];
        float o = (v > 0.f) ? v : expm1f(v);   // ELU
        out[(((size_t)b * C_ + c) * N_ + i) * L_ + l] = o;
    }
}

// ---------------------------------------------------------------------------
extern "C" void kernel_launch(void* const* d_in, const int* in_sizes, int n_in,
                              void* d_out, int out_size, void* d_ws, size_t ws_size,
                              hipStream_t stream) {
    const float* x   = (const float*)d_in[0];
    const float* adj = (const float*)d_in[1];
    const float* W   = (const float*)d_in[2];
    const float* a   = (const float*)d_in[3];
    float* out = (float*)d_out;

    // workspace (floats): Wa(512) | s1(153600) | s2(153600) | logAdj(160000)
    float* ws = (float*)d_ws;
    float* wa = ws;
    float* s1 = ws + 512;
    float* s2 = s1 + (size_t)H_ * B_ * L_ * N_;
    float* lg = s2 + (size_t)H_ * B_ * L_ * N_;

    wa_kernel<<<1, 256, 0, stream>>>(W, a, wa);
    s_kernel<<<(B_ * N_ * L_) / 256, 256, 0, stream>>>(x, wa, s1, s2);
    logadj_kernel<<<(N_ * N_ + 255) / 256, 256, 0, stream>>>(adj, lg);

    dim3 grid(N_ / 16, L_, B_);   // 25 x 12 x 8 = 2400 blocks, 4 waves each
    size_t smem = (size_t)(16 * MA_STRIDE + N_ * XS_STRIDE) * sizeof(float);
    fused_att_kernel<<<grid, 128, smem, stream>>>(x, adj, s1, s2, lg, out);
}